// GSMModell_78219944395073
// MI455X (gfx1250) — compile-verified
//
#include <hip/hip_runtime.h>

typedef __attribute__((ext_vector_type(2))) float v2f;
typedef __attribute__((ext_vector_type(4))) float v4f;
typedef __attribute__((ext_vector_type(8))) float v8f;

namespace {
constexpr int kB = 4096;
constexpr int kT = 2048;
constexpr float kInvEta = 0.1f;  // 1/ETA, ETA=10
}

// One wave32 owns TWO independent 16-row batch tiles (m0 and m0 + B/2) and
// interleaves their serial recurrences to hide the per-step dependence chain
// (fmac -> wmma -> tanh -> wmma -> wmma -> add -> fmac). All matrix math is WMMA:
//
// GEMM 1 (transposed, pre-scaled):  H = 0.5*Z^T = (0.5*W1^T) @ x^T + 0.5*b1
//   B1 = {eps, gamma} needs NO lane masking: the K=2/3 A-columns are zero.
//   sigmoid(z) = 0.5*tanh(z/2) + 0.5 -> one v_tanh_f32 per element.
// GEMM 2: de(2 x 16 bat) = (0.5*W1.*w2) @ tanh(H) + K, 16-hidden contraction
//   split into 4 K=4 WMMAs (two chained accumulator pairs) whose B operands are
//   GEMM-1 result register pairs ({th[2c],th[2c+1]} = hidden {2c,2c+1,2c+8,2c+9});
//   the "+0.5" sigmoid half folds into K[j] = 0.5*sum_h W1[j,h]*w2[h] via Cinit.
// de0 = acc[0], de1 = acc[1] land on lanes 0-15 indexed by batch, exactly where
// gamma lives -> zero cross-lane traffic. gamma provably stays 0 on lanes 16-31.
__global__ __launch_bounds__(256, 1) void gsm_scan_wmma(
    const float* __restrict__ inputs,  // [B, T, 2]
    const float* __restrict__ W1,      // [2, 16]
    const float* __restrict__ b1,      // [16]
    const float* __restrict__ W2,      // [16, 1]
    float* __restrict__ out)           // [B, T]
{
  const int lane = (int)(threadIdx.x & 31u);
  const int waveId = (int)((blockIdx.x * blockDim.x + threadIdx.x) >> 5);
  const int col = lane & 15;
  const int half = lane >> 4;
  const int hbase = half << 3;

  // ---- GEMM 1 constants (x0.5 for the tanh form) ----
  v2f A1;
  A1.x = half ? 0.0f : 0.5f * W1[col];
  A1.y = half ? 0.0f : 0.5f * W1[16 + col];
  v8f C1;
  #pragma unroll
  for (int r = 0; r < 8; ++r) C1[r] = 0.5f * b1[r + hbase];

  // ---- GEMM 2 constants: A3[c][j][k] = 0.5*W1[j][h_k]*w2[h_k] ----
  v2f A3[4];
  #pragma unroll
  for (int c = 0; c < 4; ++c) {
    float ax = 0.0f, ay = 0.0f;
    if (col < 2) {
      const int h0 = 2 * c + hbase;
      const int h1 = 2 * c + 1 + hbase;
      ax = 0.5f * W1[col * 16 + h0] * W2[h0];
      ay = 0.5f * W1[col * 16 + h1] * W2[h1];
    }
    A3[c].x = ax;
    A3[c].y = ay;
  }

  float K0 = 0.0f, K1 = 0.0f;
  #pragma unroll
  for (int h = 0; h < 16; ++h) {
    K0 += W1[h] * W2[h];
    K1 += W1[16 + h] * W2[h];
  }
  v8f Cinit = {};
  Cinit[0] = half ? 0.0f : 0.5f * K0;
  Cinit[1] = half ? 0.0f : 0.5f * K1;
  const v8f Zero = {};

  // Two independent batch tiles per wave
  const int m0a = waveId << 4;
  const int m0b = m0a + (kB >> 1);
  const float* inp[2];
  float* outp[2];
  float gamma[2] = { 0.0f, 0.0f };
  inp[0]  = inputs + (size_t)(m0a + col) * (size_t)kT * 2u;
  inp[1]  = inputs + (size_t)(m0b + col) * (size_t)kT * 2u;
  outp[0] = out    + (size_t)(m0a + col) * (size_t)kT;
  outp[1] = out    + (size_t)(m0b + col) * (size_t)kT;

  // Ping-pong input buffers: 8 steps (16 floats = 4 x b128) per tile per block
  v4f bufA[2][4], bufB[2][4];
  #pragma unroll
  for (int q = 0; q < 2; ++q)
    #pragma unroll
    for (int i = 0; i < 4; ++i) bufA[q][i] = *(const v4f*)(inp[q] + 4 * i);

  // Process 8 timesteps for both tiles from `bf`, starting at timestep tb.
  auto proc8 = [&](v4f (&bf)[2][4], int tb) __attribute__((always_inline)) {
    #pragma unroll
    for (int g = 0; g < 2; ++g) {   // two 4-step groups -> one b128 store each
      v4f so[2];
      #pragma unroll
      for (int v = 0; v < 4; ++v) {
        const int u = g * 4 + v;
        #pragma unroll
        for (int q = 0; q < 2; ++q) {
          const float eps = bf[q][u >> 1][(u & 1) * 2 + 0];
          const float hs  = bf[q][u >> 1][(u & 1) * 2 + 1];
          const float hscl = -kInvEta * hs;

          v2f B1;
          B1.x = eps;       // no masking: A's K=2 column is zero
          B1.y = gamma[q];  // no masking: gamma stays 0 on lanes 16-31

          v8f zh = __builtin_amdgcn_wmma_f32_16x16x4_f32(
              false, A1, false, B1, (short)0, C1, false, false);

          v8f tv;
          #pragma unroll
          for (int r = 0; r < 8; ++r) tv[r] = __builtin_amdgcn_tanhf(zh[r]);

          v2f Bc0; Bc0.x = tv[0]; Bc0.y = tv[1];
          v2f Bc1; Bc1.x = tv[2]; Bc1.y = tv[3];
          v2f Bc2; Bc2.x = tv[4]; Bc2.y = tv[5];
          v2f Bc3; Bc3.x = tv[6]; Bc3.y = tv[7];

          // Two chained accumulator pairs (halves live registers vs 4 parallel)
          v8f acc0 = __builtin_amdgcn_wmma_f32_16x16x4_f32(
              false, A3[0], false, Bc0, (short)0, Cinit, false, false);
          v8f acc1 = __builtin_amdgcn_wmma_f32_16x16x4_f32(
              false, A3[1], false, Bc1, (short)0, Zero, false, false);
          acc0 = __builtin_amdgcn_wmma_f32_16x16x4_f32(
              false, A3[2], false, Bc2, (short)0, acc0, false, false);
          acc1 = __builtin_amdgcn_wmma_f32_16x16x4_f32(
              false, A3[3], false, Bc3, (short)0, acc1, false, false);

          const float de0 = acc0[0] + acc1[0];
          const float de1 = acc0[1] + acc1[1];

          gamma[q] = fmaf(hscl, de1, gamma[q]);
          so[q][v] = de0;
        }
      }
      if (lane < 16) {
        *(v4f*)(outp[0] + tb + g * 4) = so[0];
        *(v4f*)(outp[1] + tb + g * 4) = so[1];
      }
    }
  };

  for (int tb = 0; tb < kT; tb += 16) {
    // Prefetch next 8 steps into B while computing from A
    #pragma unroll
    for (int q = 0; q < 2; ++q)
      #pragma unroll
      for (int i = 0; i < 4; ++i)
        bufB[q][i] = *(const v4f*)(inp[q] + (size_t)(2 * (tb + 8)) + 4 * i);

    proc8(bufA, tb);

    // Prefetch the block after next into A while computing from B
    const size_t noff = (tb + 16 < kT) ? (size_t)(2 * (tb + 16)) : 0u;
    #pragma unroll
    for (int q = 0; q < 2; ++q)
      #pragma unroll
      for (int i = 0; i < 4; ++i)
        bufA[q][i] = *(const v4f*)(inp[q] + noff + 4 * i);

    proc8(bufB, tb + 8);
  }
}

extern "C" void kernel_launch(void* const* d_in, const int* in_sizes, int n_in,
                              void* d_out, int out_size, void* d_ws, size_t ws_size,
                              hipStream_t stream) {
  (void)in_sizes; (void)n_in; (void)d_ws; (void)ws_size; (void)out_size;
  const float* inputs = (const float*)d_in[0];
  const float* W1 = (const float*)d_in[1];
  const float* b1 = (const float*)d_in[2];
  const float* W2 = (const float*)d_in[3];
  // d_in[4] = b2: unused by the recurrence (de_dx does not involve b2)
  float* out = (float*)d_out;

  const int waves = kB / 32;              // 128 waves, two 16-row tiles each
  const int block = 256;                  // 8 waves per block
  const int grid = (waves * 32) / block;  // 16 blocks
  hipLaunchKernelGGL(gsm_scan_wmma, dim3(grid), dim3(block), 0, stream,
                     inputs, W1, b1, W2, out);
}